// CovarianceLayer_36369783062600
// MI455X (gfx1250) — compile-verified
//
#include <hip/hip_runtime.h>

typedef float v2f __attribute__((ext_vector_type(2)));
typedef float v8f __attribute__((ext_vector_type(8)));

constexpr int NPTS = 1024;   // signal length
constexpr int LAGS = 128;    // L
constexpr int NEXT = NPTS + 256; // wrap-extended LDS signal (max idx used: 1263)

__launch_bounds__(256)
__global__ void cov_toeplitz_kernel(const float* __restrict__ in,
                                    float* __restrict__ out)
{
    __shared__ float sr[NEXT];
    __shared__ float si[NEXT];
    __shared__ float sc[LAGS];

    const int b   = blockIdx.x;
    const int tid = threadIdx.x;

    const float* rptr = in + (size_t)b * 2 * NPTS;
    const float* iptr = rptr + NPTS;

    // Stage signal into LDS, wrap-extended so the WMMA phase needs no modulo.
    for (int n = tid; n < NEXT; n += 256) {
        int src = n & (NPTS - 1);
        sr[n] = rptr[src];
        si[n] = iptr[src];
    }
    if (tid < LAGS) sc[tid] = 0.0f;
    __syncthreads();

    // ---- Autocorrelation via WMMA: T[q,p] = sum_n s[n+q]*s[n+16p] = C[16p-q] ----
    // K = 1024 per component, split across the 8 waves of the block.
    const int wave = tid >> 5;          // 0..7
    const int lane = tid & 31;
    const int m16  = lane & 15;         // A: row q   /  B: col p
    const int koff = (lane >> 4) * 2;   // K sub-offset per 16x4 f32 layout

    v8f acc = {};
    for (int part = 0; part < 2; ++part) {
        const float* s = part ? si : sr;
        const int base = wave * 128;    // this wave's K slice
        #pragma unroll 4
        for (int t = 0; t < 32; ++t) {
            const int n0 = base + 4 * t;
            v2f a, bm;
            // A[m16, k] = s[n0 + k + m16]
            a.x  = s[n0 + koff     + m16];
            a.y  = s[n0 + koff + 1 + m16];
            // B[k, m16] = s[n0 + k + 16*m16]
            bm.x = s[n0 + koff     + 16 * m16];
            bm.y = s[n0 + koff + 1 + 16 * m16];
            acc = __builtin_amdgcn_wmma_f32_16x16x4_f32(
                false, a, false, bm, (short)0, acc, false, false);
        }
    }

    // C/D layout: lanes 0-15 hold rows q=0..7 (vgpr j), lanes 16-31 rows q=8..15;
    // column p = lane & 15. Keep entries with d = 16p - q in [0,128).
    {
        const int p  = lane & 15;
        const int qb = (lane >> 4) * 8;
        #pragma unroll
        for (int j = 0; j < 8; ++j) {
            const int d = 16 * p - (qb + j);
            if (d >= 0 && d < LAGS)
                atomicAdd(&sc[d], acc[j] * (1.0f / 128.0f));
        }
    }
    __syncthreads();

    // ---- Toeplitz expansion (store-bandwidth bound): out[b,l,m] = sc[|l-m|] ----
    float* obase = out + (size_t)b * LAGS * LAGS;
    const int f0     = tid * 64;        // 64 consecutive outputs per thread
    const int l      = f0 >> 7;
    const int mstart = f0 & 127;
    #pragma unroll
    for (int mm = 0; mm < 64; mm += 4) {
        const int m = mstart + mm;
        int d0 = l - m;     d0 = d0 < 0 ? -d0 : d0;
        int d1 = l - m - 1; d1 = d1 < 0 ? -d1 : d1;
        int d2 = l - m - 2; d2 = d2 < 0 ? -d2 : d2;
        int d3 = l - m - 3; d3 = d3 < 0 ? -d3 : d3;
        float4 v;
        v.x = sc[d0]; v.y = sc[d1]; v.z = sc[d2]; v.w = sc[d3];
        *(float4*)(obase + l * 128 + m) = v;
    }
}

extern "C" void kernel_launch(void* const* d_in, const int* in_sizes, int n_in,
                              void* d_out, int out_size, void* d_ws, size_t ws_size,
                              hipStream_t stream)
{
    const float* in = (const float*)d_in[0];
    float* out = (float*)d_out;
    const int B = in_sizes[0] / (2 * NPTS);   // 256 batches
    cov_toeplitz_kernel<<<B, 256, 0, stream>>>(in, out);
}